// MultiQueryAttention_24764781429327
// MI455X (gfx1250) — compile-verified
//
#include <hip/hip_runtime.h>

typedef __attribute__((ext_vector_type(2))) float v2f;
typedef __attribute__((ext_vector_type(8))) float v8f;
typedef __attribute__((ext_vector_type(4))) unsigned int u32x4;
typedef __attribute__((ext_vector_type(8))) int i32x8;
typedef __attribute__((ext_vector_type(4))) int i32x4;

#define BB 2
#define LL 2048
#define HID 2048
#define NH 16
#define DH 128
#define KT 32   // k-chunk staged per TDM transfer in the GEMM

static constexpr float SCORE_SCALE = 0.0883883476483184f; // 1/sqrt(128)

// D(16x16,f32) = A(16x4,f32) * B(4x16,f32) + C  -- exact fp32 matrix op on CDNA5
__device__ __forceinline__ v8f wmma_f32(v2f a, v2f b, v8f c) {
    return __builtin_amdgcn_wmma_f32_16x16x4_f32(
        false, a, false, b, (short)0, c, false, false);
}

// Tensor Data Mover: DMA a 2-D tile (tile_d1 rows x tile_d0 f32 elements,
// row stride = stride_d0 elements) from global memory into LDS at lds_off.
// D# packing per CDNA5 ISA.
// This toolchain exposes the 6-arg builtin form.
__device__ __forceinline__ void tdm_load_2d(unsigned int lds_off, const void* gptr,
                                            unsigned int tile_d0, unsigned int tile_d1,
                                            unsigned int tensor_d0, unsigned int tensor_d1,
                                            unsigned int stride_d0) {
    unsigned long long ga = (unsigned long long)(uintptr_t)gptr;
    u32x4 g0;
    g0.x = 1u;                                            // count=1, user descriptor
    g0.y = lds_off;                                       // LDS byte address
    g0.z = (unsigned int)ga;                              // global_addr[31:0]
    g0.w = (unsigned int)((ga >> 32) & 0x1FFFFFFu) | (2u << 30); // addr[56:32] | type=2
    i32x8 g1;
    g1.s0 = (int)(2u << 16);                              // wg_mask=0, data_size=4B
    g1.s1 = (int)((tensor_d0 & 0xFFFFu) << 16);           // tensor_dim0[15:0]
    g1.s2 = (int)((tensor_d0 >> 16) | ((tensor_d1 & 0xFFFFu) << 16));
    g1.s3 = (int)((tensor_d1 >> 16) | (tile_d0 << 16));   // tile_dim0
    g1.s4 = (int)(tile_d1 & 0xFFFFu);                     // tile_dim1, tile_dim2=0
    g1.s5 = (int)stride_d0;                               // tensor_dim0_stride[31:0]
    g1.s6 = 0;                                            // stride hi, dim1_stride lo
    g1.s7 = 0;
    i32x4 gz4 = {0, 0, 0, 0};                             // 2-D tile: groups 2/3 unused
    i32x8 gz8 = {0, 0, 0, 0, 0, 0, 0, 0};
    __builtin_amdgcn_tensor_load_to_lds(g0, g1, gz4, gz4, gz8, 0);
}

__device__ __forceinline__ unsigned int lds_offset(const void* p) {
    return (unsigned int)(uintptr_t)p;   // flat LDS addr: low 32 bits = LDS offset
}

// Y[M,N] = X[M,K] @ W[N,K]^T. Block = 8 waves = 128(M) x 64(N) tile.
// K staged in KT chunks via TDM into double-buffered LDS.
__global__ __launch_bounds__(256)
void gemm_xwt_wmma(const float* __restrict__ X, const float* __restrict__ W,
                   float* __restrict__ Y, int M, int N, int K) {
    __shared__ float Xs[2][128 * KT];   // 2 x 16 KB
    __shared__ float Ws[2][64 * KT];    // 2 x 8 KB

    const int lane = threadIdx.x & 31;
    const int wave = threadIdx.x >> 5;
    const int tilesN = N >> 6;
    const int m0 = (blockIdx.x / tilesN) << 7;   // 128-row block
    const int n0 = (blockIdx.x % tilesN) << 6;   // 64-col block

    const int mrow  = lane & 15;
    const int ncol  = lane & 15;
    const int kp    = (lane >> 4) << 1;          // 0 or 2
    const int rbase = (lane >> 4) << 3;          // 0 or 8

    // preload chunk 0 (wave 0 drives the DMA engine; TDM ignores EXEC)
    if (wave == 0) {
        tdm_load_2d(lds_offset(&Xs[0][0]), X + (size_t)m0 * K, KT, 128, K, M, K);
        tdm_load_2d(lds_offset(&Ws[0][0]), W + (size_t)n0 * K, KT, 64,  K, N, K);
        __builtin_amdgcn_s_wait_tensorcnt(0);
    }
    __syncthreads();

    v8f acc0 = {}, acc1 = {}, acc2 = {}, acc3 = {};
    const int nch = K / KT;
    for (int c = 0; c < nch; ++c) {
        const int cur = c & 1;
        if (wave == 0 && c + 1 < nch) {          // prefetch next chunk into other buffer
            const int nxt = cur ^ 1;
            tdm_load_2d(lds_offset(&Xs[nxt][0]), X + (size_t)m0 * K + (size_t)(c + 1) * KT,
                        KT, 128, K, M, K);
            tdm_load_2d(lds_offset(&Ws[nxt][0]), W + (size_t)n0 * K + (size_t)(c + 1) * KT,
                        KT, 64, K, N, K);
        }

        const float* xs = &Xs[cur][(wave * 16 + mrow) * KT + kp];
        const float* w0 = &Ws[cur][( 0 + ncol) * KT + kp];
        const float* w1 = &Ws[cur][(16 + ncol) * KT + kp];
        const float* w2 = &Ws[cur][(32 + ncol) * KT + kp];
        const float* w3 = &Ws[cur][(48 + ncol) * KT + kp];
#pragma unroll
        for (int k = 0; k < KT; k += 4) {
            v2f a = *(const v2f*)(xs + k);
            acc0 = wmma_f32(a, *(const v2f*)(w0 + k), acc0);
            acc1 = wmma_f32(a, *(const v2f*)(w1 + k), acc1);
            acc2 = wmma_f32(a, *(const v2f*)(w2 + k), acc2);
            acc3 = wmma_f32(a, *(const v2f*)(w3 + k), acc3);
        }

        if (wave == 0 && c + 1 < nch) __builtin_amdgcn_s_wait_tensorcnt(0);
        __syncthreads();
    }

#pragma unroll
    for (int r = 0; r < 8; ++r) {
        float* yr = Y + (size_t)(m0 + wave * 16 + rbase + r) * N + n0 + ncol;
        yr[ 0] = acc0[r];
        yr[16] = acc1[r];
        yr[32] = acc2[r];
        yr[48] = acc3[r];
    }
}

// Flash-style MQA attention: one wave per (b, h, 16-query tile).
// Q tile TDM'd to LDS once; K/V tiles TDM double-buffered (hidden behind compute).
__global__ __launch_bounds__(32)
void mqa_flash_wmma(const float* __restrict__ Q, const float* __restrict__ Km,
                    const float* __restrict__ Vm, const int* __restrict__ amask,
                    float* __restrict__ O) {
    __shared__ float Qs[16 * DH];        // 8 KB
    __shared__ float Ks[2][16 * DH];     // 16 KB
    __shared__ float Vs[2][16 * DH];     // 16 KB
    __shared__ float Ps[16 * 16];        // 1 KB: C-layout -> A-fragment bounce

    const int lane  = threadIdx.x;
    const int qtiles = LL / 16;
    const int qt = blockIdx.x % qtiles;
    const int h  = (blockIdx.x / qtiles) % NH;
    const int b  =  blockIdx.x / (qtiles * NH);
    const int q0 = qt * 16;

    const int mrow  = lane & 15;
    const int ncol  = lane & 15;
    const int kp    = (lane >> 4) << 1;
    const int rbase = (lane >> 4) << 3;

    // Q tile: 16 rows x 128 cols out of [B*L, HID]; K/V tiles contiguous 16x128
    tdm_load_2d(lds_offset(&Qs[0]), Q + (size_t)(b * LL + q0) * HID + h * DH,
                DH, 16, HID, BB * LL, HID);
    tdm_load_2d(lds_offset(&Ks[0][0]), Km + (size_t)(b * LL) * DH, DH, 16, DH, BB * LL, DH);
    tdm_load_2d(lds_offset(&Vs[0][0]), Vm + (size_t)(b * LL) * DH, DH, 16, DH, BB * LL, DH);
    __builtin_amdgcn_s_wait_tensorcnt(0);

    v8f o[8];
#pragma unroll
    for (int t = 0; t < 8; ++t) o[t] = (v8f){};
    float mi[8], li[8];
#pragma unroll
    for (int r = 0; r < 8; ++r) { mi[r] = -1e30f; li[r] = 0.0f; }

    for (int jt = 0; jt <= qt; ++jt) {
        const int cur = jt & 1;
        const int j0 = jt * 16;
        const bool more = (jt + 1 <= qt);
        if (more) {   // prefetch next K/V tiles while we compute this one
            tdm_load_2d(lds_offset(&Ks[cur ^ 1][0]),
                        Km + (size_t)(b * LL + j0 + 16) * DH, DH, 16, DH, BB * LL, DH);
            tdm_load_2d(lds_offset(&Vs[cur ^ 1][0]),
                        Vm + (size_t)(b * LL + j0 + 16) * DH, DH, 16, DH, BB * LL, DH);
        }

        // S = Q K^T : A-frag from Qs, B-frag from Ks (b.x = K[j0+n][d], contiguous d)
        v8f s = {};
#pragma unroll
        for (int kk = 0; kk < 32; ++kk) {
            v2f a = *(const v2f*)&Qs[mrow * DH + kk * 4 + kp];
            v2f bv = *(const v2f*)&Ks[cur][ncol * DH + kk * 4 + kp];
            s = wmma_f32(a, bv, s);
        }

        // scale + causal + padding mask (finite sentinel avoids inf-inf NaN)
        const int keyok = amask[b * LL + j0 + ncol];
#pragma unroll
        for (int r = 0; r < 8; ++r) {
            float sv = s[r] * SCORE_SCALE;
            const int qrow = q0 + rbase + r;
            if ((j0 + ncol > qrow) || (keyok == 0)) sv = -1e30f;
            s[r] = sv;
        }

        // online softmax; rows rbase..rbase+7 reduce within their 16-lane half
        float corr[8];
#pragma unroll
        for (int r = 0; r < 8; ++r) {
            float mx = s[r];
#pragma unroll
            for (int off = 1; off < 16; off <<= 1)
                mx = fmaxf(mx, __shfl_xor(mx, off, 32));
            const float mnew = fmaxf(mi[r], mx);
            const float c = __expf(mi[r] - mnew);
            const float p = __expf(s[r] - mnew);
            s[r] = p;
            float rs = p;
#pragma unroll
            for (int off = 1; off < 16; off <<= 1)
                rs += __shfl_xor(rs, off, 32);
            li[r] = li[r] * c + rs;
            mi[r] = mnew;
            corr[r] = c;
        }
#pragma unroll
        for (int t = 0; t < 8; ++t)
#pragma unroll
            for (int r = 0; r < 8; ++r) o[t][r] *= corr[r];

        // P: C/D layout -> LDS -> A-fragment layout
#pragma unroll
        for (int r = 0; r < 8; ++r)
            Ps[(rbase + r) * 16 + ncol] = s[r];
        __syncthreads();

        // O += P V  (B-frag of V from LDS: two rows, same column)
#pragma unroll
        for (int kk = 0; kk < 4; ++kk) {
            v2f a;
            a.x = Ps[mrow * 16 + kk * 4 + kp];
            a.y = Ps[mrow * 16 + kk * 4 + kp + 1];
#pragma unroll
            for (int t = 0; t < 8; ++t) {
                v2f bv;
                bv.x = Vs[cur][(kk * 4 + kp)     * DH + t * 16 + ncol];
                bv.y = Vs[cur][(kk * 4 + kp + 1) * DH + t * 16 + ncol];
                o[t] = wmma_f32(a, bv, o[t]);
            }
        }
        __syncthreads();

        if (more) __builtin_amdgcn_s_wait_tensorcnt(0);  // next K/V tiles landed
    }

    // normalize and write out in [B, L, H*D] layout (transpose-merge fused)
#pragma unroll
    for (int r = 0; r < 8; ++r) li[r] = 1.0f / li[r];
    float* op = O + (size_t)(b * LL + q0) * HID + h * DH;
#pragma unroll
    for (int t = 0; t < 8; ++t)
#pragma unroll
        for (int r = 0; r < 8; ++r)
            op[(size_t)(rbase + r) * HID + t * 16 + ncol] = o[t][r] * li[r];
}

extern "C" void kernel_launch(void* const* d_in, const int* in_sizes, int n_in,
                              void* d_out, int out_size, void* d_ws, size_t ws_size,
                              hipStream_t stream) {
    const float* x     = (const float*)d_in[0];
    const int*   amask = (const int*)  d_in[1];
    const float* Wq    = (const float*)d_in[2];
    const float* Wk    = (const float*)d_in[3];
    const float* Wv    = (const float*)d_in[4];
    const float* Wo    = (const float*)d_in[5];
    float* out = (float*)d_out;

    const int M = BB * LL;                        // 4096 rows
    float* Qb = (float*)d_ws;                     // [4096, 2048]
    float* Kb = Qb + (size_t)M * HID;             // [4096, 128]
    float* Vb = Kb + (size_t)M * DH;              // [4096, 128]
    float* Ab = Vb + (size_t)M * DH;              // [4096, 2048]

    const int blkBig   = (M / 128) * (HID / 64);  // 1024 blocks
    const int blkSmall = (M / 128) * (DH  / 64);  // 64 blocks

    gemm_xwt_wmma<<<blkBig,   256, 0, stream>>>(x,  Wq, Qb, M, HID, HID);
    gemm_xwt_wmma<<<blkSmall, 256, 0, stream>>>(x,  Wk, Kb, M, DH,  HID);
    gemm_xwt_wmma<<<blkSmall, 256, 0, stream>>>(x,  Wv, Vb, M, DH,  HID);

    mqa_flash_wmma<<<BB * NH * (LL / 16), 32, 0, stream>>>(Qb, Kb, Vb, amask, Ab);

    gemm_xwt_wmma<<<blkBig,   256, 0, stream>>>(Ab, Wo, out, M, HID, HID);
}